// CrossAssetGNN_68247030333751
// MI455X (gfx1250) — compile-verified
//
#include <hip/hip_runtime.h>
#include <math.h>

#define NN    50000
#define NE    1600000
#define INF_  128
#define OUTF  32
#define NH    4
#define OUTT  128      /* NH*OUTF */
#define SLOPE 0.2f
#define LDSS  132      /* padded LDS row stride (floats): (132*m)%64 banks distinct */

typedef float v2f __attribute__((ext_vector_type(2)));
typedef float v8f __attribute__((ext_vector_type(8)));
typedef int   v4i __attribute__((ext_vector_type(4)));
typedef __attribute__((address_space(1))) v4i* gv4i_p;   // global int4*
typedef __attribute__((address_space(3))) v4i* lv4i_p;   // LDS int4*

__device__ __forceinline__ void atomAddF(float* p, float v) {
  // hardware global_atomic_add_f32 (never a CAS loop)
  unsafeAtomicAdd(p, v);
}
// order-preserving float -> uint key for atomicMax over signed floats
__device__ __forceinline__ unsigned f2key(float f) {
  int i = __float_as_int(f);
  return (i >= 0) ? ((unsigned)i | 0x80000000u) : ~(unsigned)i;
}
__device__ __forceinline__ float key2f(unsigned u) {
  return (u & 0x80000000u) ? __int_as_float((int)(u & 0x7fffffffu))
                           : __int_as_float((int)~u);
}

// ---------------- kernel 1: init accumulators + transpose W ----------------
__global__ __launch_bounds__(256) void gat_zero_k(float* __restrict__ out,
                                                  float* __restrict__ denom,
                                                  unsigned* __restrict__ mc,
                                                  const float* __restrict__ W,
                                                  float* __restrict__ wt) {
  size_t t = (size_t)blockIdx.x * 256 + threadIdx.x;
  if (t < (size_t)NN * OUTT) out[t] = 0.0f;
  if (t < (size_t)NH * NN)   denom[t] = 0.0f;
  if (t < (size_t)OUTT * INF_) {
    // Wt[col][k] = W[col>>5][k][col&31]
    const int col = (int)(t >> 7), k = (int)(t & 127);
    wt[t] = W[(size_t)(col >> 5) * (INF_ * OUTF) + (size_t)k * OUTF + (col & 31)];
  }
  if (t == 0) *mc = 0u;
}

// ---------------- kernel 2: h = x @ Wcat via V_WMMA_F32_16X16X4_F32 ----------
// block = 16 nodes x 128 cols; 8 waves, each owns one 16x16 tile, K-loop 32x (K=4)
__global__ __launch_bounds__(256) void gat_gemm_k(const float* __restrict__ x,
                                                  const float* __restrict__ wt,
                                                  float* __restrict__ h) {
  __shared__ float xs[16 * LDSS];
  const int node0 = blockIdx.x * 16;
  const int tid = threadIdx.x;

  // stage 16x128 x-tile into LDS via async copy (ASYNCcnt path, no VGPR bounce)
#if __has_builtin(__builtin_amdgcn_global_load_async_to_lds_b128)
  for (int i = tid; i < 16 * 32; i += 256) {
    const int row = i >> 5, c4 = i & 31;
    const float* gp = x + (size_t)(node0 + row) * INF_ + c4 * 4;
    float* lp = xs + row * LDSS + c4 * 4;
    __builtin_amdgcn_global_load_async_to_lds_b128(
        (gv4i_p)const_cast<float*>(gp), (lv4i_p)lp, 0, 0);
  }
#if __has_builtin(__builtin_amdgcn_s_wait_asynccnt)
  __builtin_amdgcn_s_wait_asynccnt(0);
#else
  asm volatile("s_wait_asynccnt 0" ::: "memory");
#endif
#else
  for (int i = tid; i < 16 * 32; i += 256) {
    const int row = i >> 5, c4 = i & 31;
    const float4 v = *(const float4*)(x + (size_t)(node0 + row) * INF_ + c4 * 4);
    float* dst = xs + row * LDSS + c4 * 4;
    dst[0] = v.x; dst[1] = v.y; dst[2] = v.z; dst[3] = v.w;
  }
#endif
  __syncthreads();

  const int wave = tid >> 5;            // 0..7 -> 16-col tile
  const int lane = tid & 31;
  const int m    = lane & 15;           // row within half / col within tile
  const int hi   = lane >> 4;           // 0: K pair {0,1}, rows 0-7 ; 1: K pair {2,3}, rows 8-15
  const int col  = wave * 16 + m;       // 0..127  (= head*32 + o)
  const float* Wtc = wt + (size_t)col * INF_;   // transposed: contiguous in k

  v8f c = {};
#pragma unroll
  for (int kk = 0; kk < 32; ++kk) {
    const int k0 = kk * 4 + hi * 2;
    v2f a, b;
    a.x = xs[m * LDSS + k0];
    a.y = xs[m * LDSS + k0 + 1];
    b = *(const v2f*)(Wtc + k0);        // one 8B load per K-step
    c = __builtin_amdgcn_wmma_f32_16x16x4_f32(false, a, false, b, (short)0, c,
                                              false, false);
  }
  // C/D layout: VGPR r, lanes 0-15 -> M=r ; lanes 16-31 -> M=r+8 ; N = lane%16
#pragma unroll
  for (int r = 0; r < 8; ++r) {
    const int row = r + hi * 8;
    h[(size_t)(node0 + row) * OUTT + col] = c[r];
  }
}

// ---------------- kernel 3: per-node attention scalars ----------------
__global__ __launch_bounds__(256) void gat_attn_k(const float* __restrict__ h,
                                                  const float* __restrict__ avs,
                                                  const float* __restrict__ avd,
                                                  float* __restrict__ asn,
                                                  float* __restrict__ adn) {
  const int t = blockIdx.x * 256 + threadIdx.x;
  if (t >= NN * NH) return;
  const int n = t >> 2, hd = t & 3;
  const float* hp = h + (size_t)n * OUTT + hd * OUTF;
  const float* as = avs + hd * OUTF;   // a_src[H,O,1]
  const float* ad = avd + hd * OUTF;
  float s = 0.0f, d = 0.0f;
#pragma unroll
  for (int oo = 0; oo < OUTF; ++oo) {
    const float v = hp[oo];
    s = fmaf(v, as[oo], s);
    d = fmaf(v, ad[oo], d);
  }
  asn[hd * NN + n] = s;
  adn[hd * NN + n] = d;
}

// ---------------- kernel 4: edge logits + global max ----------------
__global__ __launch_bounds__(256) void gat_logit_k(const int* __restrict__ ei,
                                                   const float* __restrict__ ew,
                                                   const float* __restrict__ asn,
                                                   const float* __restrict__ adn,
                                                   float* __restrict__ earr,
                                                   unsigned* __restrict__ mc) {
  const int e = blockIdx.x * 256 + threadIdx.x;
  float mx = -3.402823466e38f;
  if (e < NE) {
    const int s = ei[e], d = ei[NE + e];
    const float w = ew[e];
#pragma unroll
    for (int hd = 0; hd < NH; ++hd) {
      float v = asn[hd * NN + s] + adn[hd * NN + d];
      v = (v >= 0.0f) ? v : SLOPE * v;   // leaky relu
      v *= w;
      earr[(size_t)hd * NE + e] = v;
      mx = fmaxf(mx, v);
    }
  }
#pragma unroll
  for (int off = 16; off > 0; off >>= 1) mx = fmaxf(mx, __shfl_xor(mx, off, 32));
  __shared__ float smax[8];
  const int lane = threadIdx.x & 31, wv = threadIdx.x >> 5;
  if (lane == 0) smax[wv] = mx;
  __syncthreads();
  if (threadIdx.x == 0) {
    float m = smax[0];
#pragma unroll
    for (int i = 1; i < 8; ++i) m = fmaxf(m, smax[i]);
    atomicMax(mc, f2key(m));
  }
}

// ---------------- kernel 5: exp + softmax denominator ----------------
__global__ __launch_bounds__(256) void gat_exp_k(const int* __restrict__ ei,
                                                 float* __restrict__ earr,
                                                 float* __restrict__ denom,
                                                 const unsigned* __restrict__ mc) {
  const int e = blockIdx.x * 256 + threadIdx.x;
  if (e >= NE) return;
  const float m = key2f(*mc);
  const int d = ei[NE + e];
#pragma unroll
  for (int hd = 0; hd < NH; ++hd) {
    const float p = __expf(earr[(size_t)hd * NE + e] - m);
    earr[(size_t)hd * NE + e] = p;
    atomAddF(&denom[hd * NN + d], p);
  }
}

// ---------------- kernel 6: gather h[src]*alpha, scatter to out[dst] --------
// one wave per edge; lane covers 4 consecutive cols (head = lane/8)
__global__ __launch_bounds__(256) void gat_agg_k(const int* __restrict__ ei,
                                                 const float* __restrict__ earr,
                                                 const float* __restrict__ denom,
                                                 const float* __restrict__ h,
                                                 float* __restrict__ out) {
  const size_t tid = (size_t)blockIdx.x * 256 + threadIdx.x;
  const int e = (int)(tid >> 5);
  if (e >= NE) return;
  const int lane = threadIdx.x & 31;
  const int s = ei[e], d = ei[NE + e];
  const int hd = lane >> 3;
  const float alpha = earr[(size_t)hd * NE + e] / (denom[hd * NN + d] + 1e-10f);
  const float4 hv = *(const float4*)(h + (size_t)s * OUTT + lane * 4);
  float* op = out + (size_t)d * OUTT + lane * 4;
  atomAddF(op + 0, hv.x * alpha);
  atomAddF(op + 1, hv.y * alpha);
  atomAddF(op + 2, hv.z * alpha);
  atomAddF(op + 3, hv.w * alpha);
}

extern "C" void kernel_launch(void* const* d_in, const int* in_sizes, int n_in,
                              void* d_out, int out_size, void* d_ws, size_t ws_size,
                              hipStream_t stream) {
  const float* x    = (const float*)d_in[0];   // [N,128]
  const int*   ei   = (const int*)d_in[1];     // [2,E]
  const float* ew   = (const float*)d_in[2];   // [E]
  const float* W    = (const float*)d_in[3];   // [H,128,32]
  const float* avs  = (const float*)d_in[4];   // [H,32,1]
  const float* avd  = (const float*)d_in[5];   // [H,32,1]
  float* out = (float*)d_out;                  // [N,128]

  // workspace layout (bytes)
  char* ws = (char*)d_ws;
  float*    hbuf  = (float*)(ws);                    // 25,600,000  h[N,128]
  float*    earr  = (float*)(ws + 25600000);         // 25,600,000  e/e_exp [H,E]
  float*    asn   = (float*)(ws + 51200000);         //    800,000  attn_src [H,N]
  float*    adn   = (float*)(ws + 52000000);         //    800,000  attn_dst [H,N]
  float*    denom = (float*)(ws + 52800000);         //    800,000  denom [H,N]
  unsigned* mc    = (unsigned*)(ws + 53600000);      //          4  global max key
  float*    wt    = (float*)(ws + 53600256);         //     65,536  Wt[128,128] (transposed)

  // 1) zero accumulators + build transposed W (d_out is poisoned; re-init every call)
  gat_zero_k<<<25000, 256, 0, stream>>>(out, denom, mc, W, wt);
  // 2) fp32 WMMA GEMM: h = x @ Wcat   (50000/16 = 3125 row tiles)
  gat_gemm_k<<<3125, 256, 0, stream>>>(x, wt, hbuf);
  // 3) per-node attention scalars
  gat_attn_k<<<(NN * NH + 255) / 256, 256, 0, stream>>>(hbuf, avs, avd, asn, adn);
  // 4) edge logits + global max
  gat_logit_k<<<NE / 256, 256, 0, stream>>>(ei, ew, asn, adn, earr, mc);
  // 5) stabilized exp + segment-sum denominator
  gat_exp_k<<<NE / 256, 256, 0, stream>>>(ei, earr, denom, mc);
  // 6) weighted message aggregation (wave per edge)
  gat_agg_k<<<NE / 8, 256, 0, stream>>>(ei, earr, denom, hbuf, out);
}